// child_r_2456721293623
// MI455X (gfx1250) — compile-verified
//
#include <hip/hip_runtime.h>

#define N_NODE 8192
#define H1     512
#define R_EMB  128
#define N_EDGE 262144
#define N_REG  2048

typedef __attribute__((ext_vector_type(2))) float v2f;
typedef __attribute__((ext_vector_type(8))) float v8f;

// ---------------------------------------------------------------- degree/norm
__global__ void k_deg_init(float* __restrict__ deg) {
    int v = blockIdx.x * blockDim.x + threadIdx.x;
    if (v < N_NODE) deg[v] = 1.0f;            // self-loop contribution
}

__global__ void k_deg_edges(const int* __restrict__ ei, float* __restrict__ deg) {
    int e = blockIdx.x * blockDim.x + threadIdx.x;
    if (e < N_EDGE) atomicAdd(&deg[ei[2 * e + 1]], 1.0f);
}

__global__ void k_dis(const float* __restrict__ deg, float* __restrict__ dis) {
    int v = blockIdx.x * blockDim.x + threadIdx.x;
    if (v < N_NODE) dis[v] = rsqrtf(deg[v]);  // deg >= 1 always (self loop)
}

// ------------------------------------------------- layer 1: init + edge scatter
// h[v,:] = dis[v]^2 * W1[v,:] + b1
__global__ void k_h1_init(const float* __restrict__ dis, const float* __restrict__ W1,
                          const float* __restrict__ b1, float* __restrict__ h) {
    int i = blockIdx.x * blockDim.x + threadIdx.x;          // N_NODE*H1/4 threads
    if (i >= N_NODE * (H1 / 4)) return;
    int v  = i / (H1 / 4);
    int c4 = (i % (H1 / 4)) * 4;
    float s = dis[v]; float n = s * s;
    const float4 w  = *(const float4*)(W1 + (size_t)v * H1 + c4);
    const float4 bb = *(const float4*)(b1 + c4);
    float4 r; r.x = w.x * n + bb.x; r.y = w.y * n + bb.y;
    r.z = w.z * n + bb.z; r.w = w.w * n + bb.w;
    *(float4*)(h + (size_t)v * H1 + c4) = r;
}

// h[dst,:] += W1[src,:] * dis[src]*dis[dst]   (float4 gather, 4x f32 atomics)
__global__ void k_scatter1(const int* __restrict__ ei, const float* __restrict__ dis,
                           const float* __restrict__ W1, float* __restrict__ h) {
    int t = blockIdx.x * blockDim.x + threadIdx.x;          // N_EDGE * 128 threads
    int e  = t >> 7;
    int c4 = (t & 127) << 2;
    if (e >= N_EDGE) return;
    int s = ei[2 * e], d = ei[2 * e + 1];
    float norm = dis[s] * dis[d];
    const float4 w = *(const float4*)(W1 + (size_t)s * H1 + c4);
    float* o = h + (size_t)d * H1 + c4;
    atomicAdd(o + 0, w.x * norm);
    atomicAdd(o + 1, w.y * norm);
    atomicAdd(o + 2, w.z * norm);
    atomicAdd(o + 3, w.w * norm);
}

// ------------------------------------------------- layer 2 GEMM: h2 = relu(h) @ W2
// fp32 WMMA 16x16x4; block = 8 waves; block tile = 128(M) x 16(N); K = 512.
// B panel (512 x 16) of W2 staged in LDS, [k][n] layout -> conflict-free.
__global__ void __launch_bounds__(256)
k_gemm_wmma(const float* __restrict__ h, const float* __restrict__ W2,
            float* __restrict__ h2) {
    __shared__ float Bs[H1 * 16];                            // 32 KB
    const int tid = threadIdx.x;
    const int n0  = blockIdx.y * 16;
    for (int i = tid; i < H1 * 16; i += 256) {
        int k = i >> 4, c = i & 15;
        Bs[i] = W2[(size_t)k * R_EMB + n0 + c];
    }
    __syncthreads();

    const int wave = tid >> 5;
    const int lane = tid & 31;
    const int lo = lane & 15;                                // M row / N col / B col
    const int hi = lane >> 4;                                // K-half select
    const int m0 = blockIdx.x * 128 + wave * 16;

    const float* aptr = h + (size_t)(m0 + lo) * H1 + 2 * hi; // K = k + 2*hi, +1
    v8f acc = {};
    #pragma unroll 8
    for (int k = 0; k < H1; k += 4) {
        float2 av = *(const float2*)(aptr + k);
        v2f a; a.x = fmaxf(av.x, 0.0f); a.y = fmaxf(av.y, 0.0f);   // fused ReLU
        v2f b; b.x = Bs[(k + hi) * 16 + lo];                        // K = k + hi
               b.y = Bs[(k + 2 + hi) * 16 + lo];                    // K = k + 2 + hi
        acc = __builtin_amdgcn_wmma_f32_16x16x4_f32(
                  false, a, false, b, (short)0, acc, false, false);
    }
    #pragma unroll
    for (int r = 0; r < 8; ++r)                              // M = r + 8*hi, N = lo
        h2[(size_t)(m0 + r + 8 * hi) * R_EMB + n0 + lo] = acc[r];
}

// ------------------------------------------------- layer 2: init + edge scatter
__global__ void k_out2_init(const float* __restrict__ dis, const float* __restrict__ h2,
                            const float* __restrict__ b2, float* __restrict__ out2) {
    int i = blockIdx.x * blockDim.x + threadIdx.x;           // N_NODE*R_EMB/4
    if (i >= N_NODE * (R_EMB / 4)) return;
    int v  = i / (R_EMB / 4);
    int c4 = (i % (R_EMB / 4)) * 4;
    float s = dis[v]; float n = s * s;
    const float4 w  = *(const float4*)(h2 + (size_t)v * R_EMB + c4);
    const float4 bb = *(const float4*)(b2 + c4);
    float4 r; r.x = w.x * n + bb.x; r.y = w.y * n + bb.y;
    r.z = w.z * n + bb.z; r.w = w.w * n + bb.w;
    *(float4*)(out2 + (size_t)v * R_EMB + c4) = r;
}

__global__ void k_scatter2(const int* __restrict__ ei, const float* __restrict__ dis,
                           const float* __restrict__ h2, float* __restrict__ out2) {
    int t = blockIdx.x * blockDim.x + threadIdx.x;           // N_EDGE * 32 threads
    int e  = t >> 5;
    int c4 = (t & 31) << 2;
    if (e >= N_EDGE) return;
    int s = ei[2 * e], d = ei[2 * e + 1];
    float norm = dis[s] * dis[d];
    const float4 w = *(const float4*)(h2 + (size_t)s * R_EMB + c4);
    float* o = out2 + (size_t)d * R_EMB + c4;
    atomicAdd(o + 0, w.x * norm);
    atomicAdd(o + 1, w.y * norm);
    atomicAdd(o + 2, w.z * norm);
    atomicAdd(o + 3, w.w * norm);
}

__global__ void k_gather(const int* __restrict__ reg_id, const float* __restrict__ out2,
                         float* __restrict__ out) {
    int i = blockIdx.x * blockDim.x + threadIdx.x;           // N_REG * R_EMB
    if (i >= N_REG * R_EMB) return;
    int r = i >> 7, c = i & 127;
    out[i] = out2[(size_t)reg_id[r] * R_EMB + c];
}

// ----------------------------------------------------------------- launcher
extern "C" void kernel_launch(void* const* d_in, const int* in_sizes, int n_in,
                              void* d_out, int out_size, void* d_ws, size_t ws_size,
                              hipStream_t stream) {
    const int*   reg_id = (const int*)  d_in[0];
    const int*   ei     = (const int*)  d_in[1];
    const float* W1     = (const float*)d_in[2];
    const float* b1     = (const float*)d_in[3];
    const float* W2     = (const float*)d_in[4];
    const float* b2     = (const float*)d_in[5];
    float* out = (float*)d_out;

    float* ws   = (float*)d_ws;
    float* deg  = ws;                                   //  8192
    float* dis  = ws + N_NODE;                          //  8192
    float* h    = ws + 2 * N_NODE;                      //  8192*512
    float* h2   = h  + (size_t)N_NODE * H1;             //  8192*128
    float* out2 = h2 + (size_t)N_NODE * R_EMB;          //  8192*128

    const int B = 256;
    k_deg_init <<<(N_NODE + B - 1) / B, B, 0, stream>>>(deg);
    k_deg_edges<<<(N_EDGE + B - 1) / B, B, 0, stream>>>(ei, deg);
    k_dis      <<<(N_NODE + B - 1) / B, B, 0, stream>>>(deg, dis);

    k_h1_init  <<<(N_NODE * (H1 / 4) + B - 1) / B, B, 0, stream>>>(dis, W1, b1, h);
    k_scatter1 <<<(N_EDGE * 128) / B, B, 0, stream>>>(ei, dis, W1, h);

    k_gemm_wmma<<<dim3(N_NODE / 128, R_EMB / 16), B, 0, stream>>>(h, W2, h2);

    k_out2_init<<<(N_NODE * (R_EMB / 4) + B - 1) / B, B, 0, stream>>>(dis, h2, b2, out2);
    k_scatter2 <<<(N_EDGE * 32) / B, B, 0, stream>>>(ei, dis, h2, out2);

    k_gather   <<<(N_REG * R_EMB + B - 1) / B, B, 0, stream>>>(reg_id, out2, out);
}